// EncoderBlock_91259465105884
// MI455X (gfx1250) — compile-verified
//
#include <hip/hip_runtime.h>
#include <math.h>
#include <stdint.h>

// Model dims (fixed by the reference)
#define B_N  64
#define CIN_N 8
#define H_N  128
#define L_N  4096
#define NL_N 2
#define N2_N 64
#define EPS_F 1e-5f

typedef __attribute__((ext_vector_type(2))) float v2f;
typedef __attribute__((ext_vector_type(8))) float v8f;
typedef __attribute__((ext_vector_type(4))) unsigned u32x4;
typedef __attribute__((ext_vector_type(8))) unsigned u32x8;

__device__ __forceinline__ float gelu_exact(float x) {
    return 0.5f * x * (1.0f + erff(x * 0.70710678118654752f));
}

// Low 32 bits of a flat pointer to a __shared__ object == LDS byte address
// (FLAT->LDS aperture mapping truncates to addr[31:0]).
__device__ __forceinline__ unsigned lds_addr_of(const void* p) {
    return (unsigned)(uintptr_t)p;
}

// ---------------------------------------------------------------------------
// 1) Conv1d(k=3, pad=1, no bias): x (B,8,L) -> h (B,128,L)
// ---------------------------------------------------------------------------
__global__ __launch_bounds__(256) void conv1d_kernel(const float* __restrict__ x,
                                                     const float* __restrict__ w,
                                                     float* __restrict__ h) {
    __shared__ float xs[CIN_N][260];          // window [l0-1, l0+256]
    __shared__ float ws[H_N * CIN_N * 3];     // 12 KB of weights
    const int tiles = L_N / 256;
    const int b  = blockIdx.x / tiles;
    const int l0 = (blockIdx.x % tiles) * 256;
    const int t  = threadIdx.x;

    for (int i = t; i < H_N * CIN_N * 3; i += 256) ws[i] = w[i];
    for (int i = t; i < CIN_N * 258; i += 256) {
        int c = i / 258, j = i % 258;
        int l = l0 - 1 + j;
        float v = 0.0f;
        if (l >= 0 && l < L_N) v = x[(b * CIN_N + c) * L_N + l];
        xs[c][j] = v;
    }
    __syncthreads();

    for (int o = 0; o < H_N; ++o) {
        float acc = 0.0f;
#pragma unroll
        for (int c = 0; c < CIN_N; ++c) {
            const float* wo = &ws[(o * CIN_N + c) * 3];
            acc += xs[c][t]     * wo[0];
            acc += xs[c][t + 1] * wo[1];
            acc += xs[c][t + 2] * wo[2];
        }
        h[((long long)b * H_N + o) * L_N + l0 + t] = acc;
    }
}

// ---------------------------------------------------------------------------
// 2) BatchNorm statistics: one block per channel, fold into scale/shift
// ---------------------------------------------------------------------------
__global__ __launch_bounds__(256) void bn_stats_kernel(const float* __restrict__ h,
                                                       const float* __restrict__ gamma,
                                                       const float* __restrict__ beta,
                                                       float* __restrict__ scale,
                                                       float* __restrict__ shift) {
    __shared__ float s1[256], s2[256];
    const int o = blockIdx.x;
    float sum = 0.0f, sq = 0.0f;
    for (int b = 0; b < B_N; ++b) {
        const float* p = h + ((long long)b * H_N + o) * L_N;
        for (int l = threadIdx.x; l < L_N; l += 256) {
            float v = p[l];
            sum += v; sq += v * v;
        }
    }
    s1[threadIdx.x] = sum; s2[threadIdx.x] = sq;
    __syncthreads();
    for (int st = 128; st > 0; st >>= 1) {
        if (threadIdx.x < st) {
            s1[threadIdx.x] += s1[threadIdx.x + st];
            s2[threadIdx.x] += s2[threadIdx.x + st];
        }
        __syncthreads();
    }
    if (threadIdx.x == 0) {
        const float inv = 1.0f / (float)(B_N * L_N);
        float mean = s1[0] * inv;
        float var  = s2[0] * inv - mean * mean;
        float rstd = rsqrtf(var + EPS_F);
        float sc = gamma[o] * rstd;
        scale[o] = sc;
        shift[o] = beta[o] - mean * sc;
    }
}

// ---------------------------------------------------------------------------
// 3) Apply BN + exact GELU in place on h
// ---------------------------------------------------------------------------
__global__ __launch_bounds__(256) void bn_gelu_kernel(float* __restrict__ h,
                                                      const float* __restrict__ scale,
                                                      const float* __restrict__ shift) {
    long long idx = (long long)blockIdx.x * 256 + threadIdx.x;
    int c = (int)((idx / L_N) % H_N);
    float v = h[idx] * scale[c] + shift[c];
    h[idx] = gelu_exact(v);
}

// ---------------------------------------------------------------------------
// 4) 128x128 GEMM over channel dim using V_WMMA_F32_16X16X4_F32.
//    Weight staged into LDS by the Tensor Data Mover (tensor_load_to_lds):
//    one TDM descriptor loads the whole 128x128 tile and hardware-pads each
//    128-DWORD row by 4 DWORDs -> LDS stride 132 (bank-conflict-free A reads).
//    Block: 256 thr (8 waves). Wave w owns 16 columns x all 8 M-tiles.
// ---------------------------------------------------------------------------
#define WLDS_STRIDE 132
__global__ __launch_bounds__(256) void gemm128_wmma_kernel(const float* __restrict__ W,
                                                           const float* __restrict__ bias,
                                                           const float* __restrict__ in,
                                                           float* __restrict__ out,
                                                           int Lc) {
    __shared__ float wl[H_N * WLDS_STRIDE];   // ~66 KB
    const int t = threadIdx.x;

    if (t < 32) {  // wave 0 issues one TDM load for the whole weight tile
        unsigned long long ga = (unsigned long long)(uintptr_t)W;
        u32x4 g0;
        g0.x = 1u;                                    // count=1 (valid D#)
        g0.y = lds_addr_of(&wl[0]);                   // lds_addr (bytes)
        g0.z = (unsigned)ga;                          // global_addr[31:0]
        g0.w = (unsigned)((ga >> 32) & 0x01FFFFFFull) // global_addr[56:32]
             | (2u << 30);                            // type=2 ("image")
        u32x8 g1;
        g1[0] = (2u << 16)     // data_size = 4 bytes
              | (1u << 20)     // pad_enable
              | (6u << 22)     // pad_interval: every 128 DWORDs (one row)
              | (3u << 25);    // pad_amount: 4 DWORDs -> stride 132
        g1[1] = (128u << 16);  // tensor_dim0 = 128 (low 16 in [31:16])
        g1[2] = (128u << 16);  // tensor_dim1 = 128
        g1[3] = (128u << 16);  // tile_dim0 = 128
        g1[4] = 128u;          // tile_dim1 = 128, tile_dim2 = 0
        g1[5] = 128u;          // tensor_dim0_stride = 128 (low 32)
        g1[6] = 0u;
        g1[7] = 0u;
        asm volatile("tensor_load_to_lds %0, %1" :: "s"(g0), "s"(g1) : "memory");
        __builtin_amdgcn_s_wait_tensorcnt(0);
    }
    __syncthreads();

    const int wave  = t >> 5;
    const int lane  = t & 31;
    const int lhalf = lane >> 4;     // 0: lanes 0-15, 1: lanes 16-31
    const int lmod  = lane & 15;

    const int tilesPerB = Lc / 128;
    const int b   = blockIdx.x / tilesPerB;
    const int col = (blockIdx.x % tilesPerB) * 128 + wave * 16 + lmod;
    const float* inb = in + (long long)b * H_N * Lc;

    v8f acc[8];
    const v8f vzero = {0.f, 0.f, 0.f, 0.f, 0.f, 0.f, 0.f, 0.f};
#pragma unroll
    for (int m = 0; m < 8; ++m) acc[m] = vzero;

    for (int k0 = 0; k0 < H_N; k0 += 4) {
        const int kb = k0 + 2 * lhalf;  // K index per ISA A/B fragment layout
        v2f bfrag;
        bfrag.x = inb[(long long)kb * Lc + col];
        bfrag.y = inb[(long long)(kb + 1) * Lc + col];
        if (k0 + 4 < H_N)   // hint the next B slab into cache (global_prefetch_b8)
            __builtin_prefetch(&inb[(long long)(kb + 4) * Lc + col], 0, 3);
#pragma unroll
        for (int m = 0; m < 8; ++m) {
            v2f afrag;
            afrag.x = wl[(m * 16 + lmod) * WLDS_STRIDE + kb];
            afrag.y = wl[(m * 16 + lmod) * WLDS_STRIDE + kb + 1];
            acc[m] = __builtin_amdgcn_wmma_f32_16x16x4_f32(
                false, afrag, false, bfrag, (short)0, acc[m], false, false);
        }
    }
    // C/D layout: VGPR v holds M = v (lanes 0-15) / M = v+8 (lanes 16-31)
#pragma unroll
    for (int m = 0; m < 8; ++m) {
#pragma unroll
        for (int v = 0; v < 8; ++v) {
            int row = m * 16 + v + 8 * lhalf;
            out[((long long)b * H_N + row) * Lc + col] = acc[m][v] + bias[row];
        }
    }
}

// ---------------------------------------------------------------------------
// 5a) S4D discretization: per (h,n) compute decay e^{dtA} and 2*C*(e^{dtA}-1)/A
// ---------------------------------------------------------------------------
__global__ __launch_bounds__(256) void s4d_coeff_kernel(const float* __restrict__ log_dt,
                                                        const float* __restrict__ A_log_real,
                                                        const float* __restrict__ A_imag,
                                                        const float* __restrict__ C_re,
                                                        const float* __restrict__ C_im,
                                                        float* __restrict__ coeff,
                                                        int layer) {
    const int idx = blockIdx.x * 256 + threadIdx.x;   // H*N2 = 8192
    const int h = idx / N2_N;
    const float dt = expf(log_dt[layer * H_N + h]);
    const int gi = layer * H_N * N2_N + idx;
    const float Ar = -expf(A_log_real[gi]);
    const float Ai = A_imag[gi];
    const float dr = Ar * dt, di = Ai * dt;
    const float em = expf(dr);
    const float er = em * cosf(di);
    const float ei = em * sinf(di);
    // (e^{dtA}-1)/A  via multiply by conj(A)/|A|^2
    const float nr = er - 1.0f, ni = ei;
    const float inv = 1.0f / (Ar * Ar + Ai * Ai);
    const float qr = (nr * Ar + ni * Ai) * inv;
    const float qi = (ni * Ar - nr * Ai) * inv;
    const float cr = C_re[gi], ci = C_im[gi];
    coeff[idx]             = er;                          // decay (re)
    coeff[8192 + idx]      = ei;                          // decay (im)
    coeff[2 * 8192 + idx]  = 2.0f * (cr * qr - ci * qi);  // 2*Ceff (re)
    coeff[3 * 8192 + idx]  = 2.0f * (cr * qi + ci * qr);  // 2*Ceff (im)
}

// ---------------------------------------------------------------------------
// 5b) S4D recurrent scan, fused with D*u skip, GELU and residual.
//     One block per (b,h); 64 lanes = 64 diagonal complex states.
//     Chunks of z stream through a DOUBLE-BUFFERED LDS stage filled with
//     global_load_async_to_lds_b64 one chunk ahead (ASYNCcnt pipelining),
//     so the fetch of chunk c+1 overlaps the 128-step recurrence of chunk c.
// ---------------------------------------------------------------------------
#define T_CH 128
#define TP   129   // padded stride -> conflict-free LDS part buffer
__global__ __launch_bounds__(64) void s4d_scan_kernel(float* __restrict__ z,
                                                      const float* __restrict__ coeff,
                                                      const float* __restrict__ Dp,
                                                      int layer) {
    __shared__ float us[2][T_CH];
    __shared__ float part[N2_N * TP];
    const int bh = blockIdx.x;            // b*H + h
    const int h  = bh % H_N;
    const int n  = threadIdx.x;           // state index 0..63
    const int ci = h * N2_N + n;
    const float ar  = coeff[ci];
    const float ai  = coeff[8192 + ci];
    const float c2r = coeff[2 * 8192 + ci];
    const float c2i = coeff[3 * 8192 + ci];
    const float Dh  = Dp[layer * H_N + h];
    float* zp = z + (long long)bh * L_N;

    // Per-thread async staging: thread n moves 8 bytes (elements 2n, 2n+1).
    const unsigned ldsA = lds_addr_of(&us[0][2 * n]);
    const unsigned ldsB = lds_addr_of(&us[1][2 * n]);

    // Prologue: chunk 0 -> buffer 0
    asm volatile("global_load_async_to_lds_b64 %0, %1, off"
                 :: "v"(ldsA), "v"(zp + 2 * n) : "memory");

    float sr = 0.0f, si = 0.0f;
    for (int c = 0; c < L_N / T_CH; ++c) {
        const int cur = c & 1;
        if (c + 1 < L_N / T_CH) {
            // issue chunk c+1 into the other buffer, then wait for chunk c
            const float* nxt = zp + (c + 1) * T_CH + 2 * n;
            asm volatile("global_load_async_to_lds_b64 %0, %1, off"
                         :: "v"(cur ? ldsA : ldsB), "v"(nxt) : "memory");
            asm volatile("s_wait_asynccnt 0x1" ::: "memory");  // in-order: c done
        } else {
            asm volatile("s_wait_asynccnt 0x0" ::: "memory");
        }
        __syncthreads();

        const float* uu = us[cur];
        float* pp = part + n * TP;
#pragma unroll 4
        for (int tt = 0; tt < T_CH; ++tt) {
            float u   = uu[tt];
            float nsr = ar * sr - ai * si + u;
            float nsi = ar * si + ai * sr;
            sr = nsr; si = nsi;
            pp[tt] = c2r * sr - c2i * si;   // Re(2*Ceff * s)
        }
        __syncthreads();

#pragma unroll
        for (int rep = 0; rep < 2; ++rep) {
            int tt = n + rep * 64;
            float accv = 0.0f;
            for (int m = 0; m < N2_N; ++m) accv += part[m * TP + tt];
            float u  = uu[tt];
            float yv = accv + Dh * u;
            zp[c * T_CH + tt] = u + gelu_exact(yv);
        }
        __syncthreads();
    }
}

// ---------------------------------------------------------------------------
// 6) LayerNorm over channel dim H for every (b,l)
// ---------------------------------------------------------------------------
__global__ __launch_bounds__(256) void ln_kernel(float* __restrict__ z,
                                                 const float* __restrict__ gamma,
                                                 const float* __restrict__ beta,
                                                 int layer) {
    const int g = blockIdx.x * 256 + threadIdx.x;   // B*L threads
    const int b = g / L_N, l = g % L_N;
    float* base = z + (long long)b * H_N * L_N + l;
    float sum = 0.0f, sq = 0.0f;
    for (int hh = 0; hh < H_N; ++hh) {
        float v = base[hh * L_N];
        sum += v; sq += v * v;
    }
    const float mean = sum * (1.0f / H_N);
    const float var  = sq * (1.0f / H_N) - mean * mean;
    const float rstd = rsqrtf(var + EPS_F);
    const float* ga = gamma + layer * H_N;
    const float* be = beta + layer * H_N;
    for (int hh = 0; hh < H_N; ++hh) {
        float v = base[hh * L_N];
        base[hh * L_N] = (v - mean) * rstd * ga[hh] + be[hh];
    }
}

// ---------------------------------------------------------------------------
// 7) AvgPool1d(2,2) BEFORE the decoder (pool commutes with the linear layer)
// ---------------------------------------------------------------------------
__global__ __launch_bounds__(256) void pool_kernel(const float* __restrict__ z,
                                                   float* __restrict__ zp) {
    long long idx = (long long)blockIdx.x * 256 + threadIdx.x;  // B*H*L/2
    zp[idx] = 0.5f * (z[2 * idx] + z[2 * idx + 1]);
}

// ---------------------------------------------------------------------------
// Host launcher
// ---------------------------------------------------------------------------
extern "C" void kernel_launch(void* const* d_in, const int* in_sizes, int n_in,
                              void* d_out, int out_size, void* d_ws, size_t ws_size,
                              hipStream_t stream) {
    const float* x          = (const float*)d_in[0];
    const float* conv_w     = (const float*)d_in[1];
    const float* bn_gamma   = (const float*)d_in[2];
    const float* bn_beta    = (const float*)d_in[3];
    const float* enc_w      = (const float*)d_in[4];
    const float* enc_b      = (const float*)d_in[5];
    const float* dec_w      = (const float*)d_in[6];
    const float* dec_b      = (const float*)d_in[7];
    const float* log_dt     = (const float*)d_in[8];
    const float* A_log_real = (const float*)d_in[9];
    const float* A_imag     = (const float*)d_in[10];
    const float* C_re       = (const float*)d_in[11];
    const float* C_im       = (const float*)d_in[12];
    const float* Dp         = (const float*)d_in[13];
    const float* ln_gamma   = (const float*)d_in[14];
    const float* ln_beta    = (const float*)d_in[15];

    // Workspace layout (floats): h[33.5M] | z[33.5M] | scale/shift[256] | coeff[32768]
    float* ws    = (float*)d_ws;
    float* h     = ws;                         // (B,128,L); reused as pooled z later
    float* z     = ws + 33554432;              // (B,128,L)
    float* scale = ws + 67108864;              // 128
    float* shift = scale + 128;                // 128
    float* coeff = ws + 67109376;              // 4*8192
    float* zpool = h;                          // alias: h is dead after encoder

    conv1d_kernel<<<B_N * (L_N / 256), 256, 0, stream>>>(x, conv_w, h);
    bn_stats_kernel<<<H_N, 256, 0, stream>>>(h, bn_gamma, bn_beta, scale, shift);
    bn_gelu_kernel<<<(B_N * H_N * L_N) / 256, 256, 0, stream>>>(h, scale, shift);
    gemm128_wmma_kernel<<<(B_N * L_N) / 128, 256, 0, stream>>>(enc_w, enc_b, h, z, L_N);

    for (int i = 0; i < NL_N; ++i) {
        s4d_coeff_kernel<<<(H_N * N2_N) / 256, 256, 0, stream>>>(
            log_dt, A_log_real, A_imag, C_re, C_im, coeff, i);
        s4d_scan_kernel<<<B_N * H_N, 64, 0, stream>>>(z, coeff, Dp, i);
        ln_kernel<<<(B_N * L_N) / 256, 256, 0, stream>>>(z, ln_gamma, ln_beta, i);
    }

    pool_kernel<<<(B_N * H_N * (L_N / 2)) / 256, 256, 0, stream>>>(z, zpool);
    gemm128_wmma_kernel<<<(B_N * (L_N / 2)) / 128, 256, 0, stream>>>(
        dec_w, dec_b, zpool, (float*)d_out, L_N / 2);
}